// ChannelAttention_4741643894861
// MI455X (gfx1250) — compile-verified
//
#include <hip/hip_runtime.h>
#include <hip/hip_bf16.h>
#include <math.h>

typedef __attribute__((ext_vector_type(2))) float v2f;
typedef __attribute__((ext_vector_type(8))) float v8f;

// Problem constants (fixed by the reference).
constexpr int Bc   = 2;
constexpr int Cc   = 128;
constexpr int HWc  = 256 * 256;   // 65536
constexpr int NGc  = 4;           // channel groups (NUM_CHANS)
constexpr int Dc   = 32;          // channels per group
constexpr float EPSc = 1e-12f;

#define WMMA_F32(A, Bf, Cacc) \
    __builtin_amdgcn_wmma_f32_16x16x4_f32(false, (A), false, (Bf), (short)0, (Cacc), false, false)

// ---------------------------------------------------------------------------
// Zero-init workspace region used by atomics.
__global__ void zero_kernel(float* __restrict__ p, int n) {
    int i = blockIdx.x * 256 + threadIdx.x;
    if (i < n) p[i] = 0.f;
}

// ---------------------------------------------------------------------------
// Phase 1: per-(b,g) 32x32 channel Gram G = X * X^T over HW, plus row sums S.
// grid = (NSLICE1=32, NG, B), block = 256 (8 waves).
// Each wave: A-frag lane layout for 16x16x4 f32: lane l holds row M = l%16,
// K = 2*(l/16) + {0,1}.  For a Gram, the B-fragment (K x N, N = row of the
// other tile) has the identical layout, so the same registers feed both.
constexpr int NSLICE1 = 32;

__global__ void gram_kernel(const float* __restrict__ x,
                            float* __restrict__ G,     // [B][NG][32][32]
                            float* __restrict__ S) {   // [B][C]
    __shared__ float sG[4 * 256 + 32];  // 4 tiles of 16x16 + 2x16 row sums
    const int tid  = threadIdx.x;
    const int wave = tid >> 5;
    const int lane = tid & 31;
    const int m    = lane & 15;
    const int half = lane >> 4;
    const int slice = blockIdx.x, g = blockIdx.y, b = blockIdx.z;

    for (int i = tid; i < 4 * 256 + 32; i += 256) sG[i] = 0.f;
    __syncthreads();

    const int nPerBlock = HWc / NSLICE1;     // 2048
    const int nPerWave  = nPerBlock / 8;     // 256
    const int n0 = slice * nPerBlock + wave * nPerWave;

    const float* row0 = x + ((size_t)b * Cc + g * Dc + m) * HWc;  // tile 0
    const float* row1 = row0 + (size_t)16 * HWc;                  // tile 1

    v8f c00 = {}, c01 = {}, c10 = {}, c11 = {};
    float s0 = 0.f, s1 = 0.f;

    #pragma unroll 4
    for (int k = 0; k < nPerWave; k += 4) {
        const int n = n0 + k + 2 * half;
        v2f a0 = *(const v2f*)(row0 + n);
        v2f a1 = *(const v2f*)(row1 + n);
        s0 += a0.x + a0.y;
        s1 += a1.x + a1.y;
        c00 = WMMA_F32(a0, a0, c00);
        c01 = WMMA_F32(a0, a1, c01);
        c10 = WMMA_F32(a1, a0, c10);
        c11 = WMMA_F32(a1, a1, c11);
    }

    // C/D layout: lane l, vgpr r -> (M = r + 8*half, N = l%16)
    #pragma unroll
    for (int r = 0; r < 8; ++r) {
        const int M = r + 8 * half;
        atomicAdd(&sG[0 * 256 + M * 16 + m], c00[r]);
        atomicAdd(&sG[1 * 256 + M * 16 + m], c01[r]);
        atomicAdd(&sG[2 * 256 + M * 16 + m], c10[r]);
        atomicAdd(&sG[3 * 256 + M * 16 + m], c11[r]);
    }
    atomicAdd(&sG[1024 + m], s0);
    atomicAdd(&sG[1024 + 16 + m], s1);
    __syncthreads();

    float* Gg = G + ((size_t)b * NGc + g) * (Dc * Dc);
    for (int i = tid; i < 1024; i += 256) {
        const int tile = i >> 8, M = (i >> 4) & 15, N = i & 15;
        const int ti = tile >> 1, tj = tile & 1;
        atomicAdd(&Gg[(ti * 16 + M) * Dc + (tj * 16 + N)], sG[i]);
    }
    if (tid < 32) atomicAdd(&S[b * Cc + g * Dc + tid], sG[1024 + tid]);
}

// ---------------------------------------------------------------------------
// Phase 2: closed-form attention from Gram + sums.
// grid = B*HEADS = 16, block = 32 (lane = row d).
__global__ void attn_kernel(const float* __restrict__ G,
                            const float* __restrict__ S,
                            const float* __restrict__ wqkv,  // [C][6]
                            const float* __restrict__ bqkv,  // [C][6]
                            const float* __restrict__ t,     // [8]
                            float* __restrict__ Mv,          // [B][8][32][32]
                            float* __restrict__ cv) {        // [B][8][32]
    const int bh = blockIdx.x;
    const int b = bh >> 3, h = bh & 7;
    const int g = h & 3, eh = h >> 2;
    const int d = threadIdx.x;  // 0..31
    const int c = g * Dc + d;
    const float N = (float)HWc;

    __shared__ float ak[32], bk[32], av[32], bv[32], nk[32], Scol[32];

    const float Sc  = S[b * Cc + c];
    const float* Gg = G + ((size_t)b * NGc + g) * (Dc * Dc);
    const float Gdd = Gg[d * Dc + d];

    const float aq = wqkv[c * 6 + eh],     bq = bqkv[c * 6 + eh];
    const float akv = wqkv[c * 6 + 2 + eh], bkv = bqkv[c * 6 + 2 + eh];
    const float avv = wqkv[c * 6 + 4 + eh], bvv = bqkv[c * 6 + 4 + eh];

    const float nq = sqrtf(fmaxf(aq * aq * Gdd + 2.f * aq * bq * Sc + bq * bq * N, 0.f));
    const float nkd = sqrtf(fmaxf(akv * akv * Gdd + 2.f * akv * bkv * Sc + bkv * bkv * N, 0.f));

    ak[d] = akv; bk[d] = bkv; av[d] = avv; bv[d] = bvv; nk[d] = nkd; Scol[d] = Sc;
    __syncthreads();

    const float tt = t[h];
    const float invnq = 1.f / fmaxf(nq, EPSc);
    const float* Grow = Gg + d * Dc;

    float row[32];
    float mx = -INFINITY;
    #pragma unroll
    for (int e = 0; e < 32; ++e) {
        float dot = aq * ak[e] * Grow[e] + aq * bk[e] * Sc
                  + bq * ak[e] * Scol[e] + bq * bk[e] * N;
        float a = dot * invnq / fmaxf(nk[e], EPSc) * tt;
        row[e] = a;
        mx = fmaxf(mx, a);
    }
    float sum = 0.f;
    #pragma unroll
    for (int e = 0; e < 32; ++e) { row[e] = __expf(row[e] - mx); sum += row[e]; }
    const float inv = 1.f / sum;

    float cconst = 0.f;
    float* Mrow = Mv + (((size_t)b * 8 + h) * Dc + d) * Dc;
    #pragma unroll
    for (int e = 0; e < 32; ++e) {
        const float A = row[e] * inv;
        Mrow[e] = A * av[e];
        cconst += A * bv[e];
    }
    cv[((size_t)b * 8 + h) * Dc + d] = cconst;
}

// ---------------------------------------------------------------------------
// Phase 2b: fold w_fus over the two experts -> per-(b,g) 32x32 mixing matrix.
// grid = B*NG = 8, block = 256.
__global__ void mix_kernel(const float* __restrict__ Mv,
                           const float* __restrict__ cv,
                           const float* __restrict__ wfus,   // [C][2]
                           const float* __restrict__ bfus,   // [C]
                           float* __restrict__ Wmix,         // [B][NG][32][32]
                           float* __restrict__ biasOut) {    // [B][C]
    const int bg = blockIdx.x;
    const int b = bg >> 2, g = bg & 3;
    const int tid = threadIdx.x;
    const float* M0 = Mv + ((size_t)b * 8 + g)     * 1024;  // h = 0*4+g
    const float* M1 = Mv + ((size_t)b * 8 + 4 + g) * 1024;  // h = 1*4+g
    float* Wo = Wmix + ((size_t)b * NGc + g) * 1024;
    for (int idx = tid; idx < 1024; idx += 256) {
        const int i = idx >> 5;
        const int c = g * Dc + i;
        Wo[idx] = wfus[c * 2 + 0] * M0[idx] + wfus[c * 2 + 1] * M1[idx];
    }
    if (tid < 32) {
        const int c = g * Dc + tid;
        biasOut[b * Cc + c] = wfus[c * 2 + 0] * cv[((size_t)b * 8 + g) * Dc + tid]
                            + wfus[c * 2 + 1] * cv[((size_t)b * 8 + 4 + g) * Dc + tid]
                            + bfus[c];
    }
}

// ---------------------------------------------------------------------------
// Phase 3: out[b, g*32+i, n] = sum_j Wmix[i][j] * x[b, g*32+j, n] + bias.
// grid = (NSLICE3=64, NG, B), block = 256 (8 waves). A-fragments (Wmix) are
// hoisted into registers and reused across the whole n-range of the wave.
constexpr int NSLICE3 = 64;

__global__ void apply_kernel(const float* __restrict__ x,
                             const float* __restrict__ Wmix,
                             const float* __restrict__ biasOut,
                             float* __restrict__ out) {
    const int tid  = threadIdx.x;
    const int wave = tid >> 5;
    const int lane = tid & 31;
    const int m    = lane & 15;
    const int half = lane >> 4;
    const int slice = blockIdx.x, g = blockIdx.y, b = blockIdx.z;

    const int nPerBlock = HWc / NSLICE3;   // 1024
    const int nPerWave  = nPerBlock / 8;   // 128
    const int nbase = slice * nPerBlock + wave * nPerWave;

    const float* W = Wmix + ((size_t)b * NGc + g) * 1024;

    v2f a0[8], a1[8];
    #pragma unroll
    for (int ks = 0; ks < 8; ++ks) {
        const int kk = 4 * ks + 2 * half;
        a0[ks] = *(const v2f*)(W + m * Dc + kk);
        a1[ks] = *(const v2f*)(W + (16 + m) * Dc + kk);
    }
    float bias0[8], bias1[8];
    const float* bb = biasOut + b * Cc + g * Dc;
    #pragma unroll
    for (int r = 0; r < 8; ++r) {
        bias0[r] = bb[r + 8 * half];
        bias1[r] = bb[16 + r + 8 * half];
    }

    const float* xg = x  + ((size_t)b * Cc + g * Dc) * HWc;
    float*       og = out + ((size_t)b * Cc + g * Dc) * HWc;

    for (int n0 = nbase; n0 < nbase + nPerWave; n0 += 16) {
        v8f c0 = {}, c1 = {};
        #pragma unroll
        for (int ks = 0; ks < 8; ++ks) {
            const int kk = 4 * ks + 2 * half;
            v2f bf;
            bf.x = xg[(size_t)kk * HWc + n0 + m];
            bf.y = xg[(size_t)(kk + 1) * HWc + n0 + m];
            c0 = WMMA_F32(a0[ks], bf, c0);
            c1 = WMMA_F32(a1[ks], bf, c1);
        }
        #pragma unroll
        for (int r = 0; r < 8; ++r) {
            og[(size_t)(r + 8 * half) * HWc + n0 + m]      = c0[r] + bias0[r];
            og[(size_t)(16 + r + 8 * half) * HWc + n0 + m] = c1[r] + bias1[r];
        }
    }
}

// ---------------------------------------------------------------------------
extern "C" void kernel_launch(void* const* d_in, const int* in_sizes, int n_in,
                              void* d_out, int out_size, void* d_ws, size_t ws_size,
                              hipStream_t stream) {
    (void)in_sizes; (void)n_in; (void)out_size; (void)ws_size;
    const float* x    = (const float*)d_in[0];
    const float* wqkv = (const float*)d_in[1];
    const float* bqkv = (const float*)d_in[2];
    const float* wfus = (const float*)d_in[3];
    const float* bfus = (const float*)d_in[4];
    const float* t    = (const float*)d_in[5];
    float* out = (float*)d_out;

    // Workspace layout (floats): G[8192] S[256] Mv[16384] cv[512] Wmix[8192] bias[256]
    float* G       = (float*)d_ws;
    float* S       = G + Bc * NGc * Dc * Dc;        // +8192
    float* Mv      = S + Bc * Cc;                   // +256
    float* cv      = Mv + Bc * 8 * Dc * Dc;         // +16384
    float* Wmix    = cv + Bc * 8 * Dc;              // +512
    float* biasOut = Wmix + Bc * NGc * Dc * Dc;     // +8192

    const int nzero = Bc * NGc * Dc * Dc + Bc * Cc; // G + S (contiguous)
    zero_kernel<<<(nzero + 255) / 256, 256, 0, stream>>>(G, nzero);
    gram_kernel<<<dim3(NSLICE1, NGc, Bc), 256, 0, stream>>>(x, G, S);
    attn_kernel<<<Bc * 8, 32, 0, stream>>>(G, S, wqkv, bqkv, t, Mv, cv);
    mix_kernel<<<Bc * NGc, 256, 0, stream>>>(Mv, cv, wfus, bfus, Wmix, biasOut);
    apply_kernel<<<dim3(NSLICE3, NGc, Bc), 256, 0, stream>>>(x, Wmix, biasOut, out);
}